// UPTRec_24962349925015
// MI455X (gfx1250) — compile-verified
//
#include <hip/hip_runtime.h>
#include <float.h>

typedef __attribute__((ext_vector_type(2))) float v2f;
typedef __attribute__((ext_vector_type(8))) float v8f;

#define BATCH 4096
#define TN    200
#define HIDN  128
#define KC    10
#define KMITR 20
#define NTILE ((TN + 15) / 16)   // 13

// One workgroup (256 threads = 8 waves) per batch row.
// LDS layout (floats):
//   Xs    : TN*HIDN   = 25600   (the h row, resident for all 20 iterations)
//   Cs    : 16*HIDN   = 2048    (centers, rows 10..15 zero padding for N=16 WMMA)
//   sums  : KC*HIDN   = 1280
//   x2    : TN        = 200
//   c2    : 16
//   cnt   : 16
//   ids   : TN (ints) = 200
__global__ __launch_bounds__(256) void uptrec_kmeans_kernel(
    const int* __restrict__ user_ids,
    const int* __restrict__ seq,
    const float* __restrict__ item_emb,   // (100001, 64)
    const float* __restrict__ user_emb,   // (100001, 64)
    const float* __restrict__ pos_emb,    // (200, 128)
    float* __restrict__ out_h,            // (4096, 200, 128)
    float* __restrict__ out_ids,          // (4096, 200) as float
    float* __restrict__ out_centers)      // (4096, 10, 128)
{
    extern __shared__ float smem[];
    float* Xs     = smem;                         // 25600
    float* Cs     = Xs + TN * HIDN;               // 2048
    float* sums_s = Cs + 16 * HIDN;               // 1280
    float* x2_s   = sums_s + KC * HIDN;           // 200
    float* c2_s   = x2_s + TN;                    // 16
    float* cnt_s  = c2_s + 16;                    // 16
    int*   ids_s  = (int*)(cnt_s + 16);           // 200

    const int b    = blockIdx.x;
    const int tid  = threadIdx.x;
    const int lane = tid & 31;
    const int wave = tid >> 5;

    const int  uid     = user_ids[b];
    const int* seq_row = seq + (size_t)b * TN;

    // ---- Phase A: build h row into LDS and global (coalesced over c) ----
    for (int i = tid; i < TN * HIDN; i += 256) {
        const int t = i >> 7;
        const int c = i & 127;
        float base;
        if (c < 64) base = item_emb[(size_t)seq_row[t] * 64 + c];
        else        base = user_emb[(size_t)uid * 64 + (c - 64)];
        const float v = base + pos_emb[i];
        Xs[i] = v;
        out_h[(size_t)b * (TN * HIDN) + i] = v;
    }
    __syncthreads();

    // ---- x2[t] = ||X[t]||^2 (computed once) ----
    if (tid < TN) {
        const float* xr = Xs + tid * HIDN;
        float s = 0.f;
        #pragma unroll 4
        for (int c = 0; c < HIDN; ++c) s += xr[c] * xr[c];
        x2_s[tid] = s;
    }
    // ---- init centers = X[:10]; zero pad rows 10..15 so N=16 WMMA is safe ----
    for (int i = tid; i < 16 * HIDN; i += 256) {
        const int r = i >> 7;
        Cs[i] = (r < KC) ? Xs[i] : 0.f;
    }
    __syncthreads();

    const int r15   = lane & 15;
    const int khalf = (lane >> 4) << 1;   // 0 for lanes 0-15, 2 for lanes 16-31
    const int kk    = lane & 15;          // candidate-center index of this lane
    const int mhi   = (lane >> 4) * 8;    // D-matrix M offset for upper half-wave

    for (int it = 0; it < KMITR; ++it) {
        // ---- c2[k] = ||C[k]||^2 ----
        if (tid < 16) {
            const float* cr = Cs + tid * HIDN;
            float s = 0.f;
            #pragma unroll 4
            for (int c = 0; c < HIDN; ++c) s += cr[c] * cr[c];
            c2_s[tid] = s;
        }
        __syncthreads();

        // ---- distances via V_WMMA_F32_16X16X4_F32 + per-row argmin ----
        // Tile loop is wave-uniform -> EXEC stays all-ones around WMMA.
        for (int tile = wave; tile < NTILE; tile += 8) {
            const int t0 = tile * 16;
            int ar = t0 + r15; if (ar > TN - 1) ar = TN - 1;  // clamp partial tile
            const float* Arow = Xs + ar  * HIDN + khalf;      // A: X rows
            const float* Brow = Cs + r15 * HIDN + khalf;      // B: center rows (=C^T cols)
            v8f acc = {};
            #pragma unroll
            for (int k0 = 0; k0 < HIDN; k0 += 4) {
                v2f a  = *(const v2f*)(Arow + k0);
                v2f bb = *(const v2f*)(Brow + k0);
                acc = __builtin_amdgcn_wmma_f32_16x16x4_f32(
                        false, a, false, bb, (short)0, acc, false, false);
            }
            // D layout: VGPR v, lane L -> row M = v + 8*(L>=16), col N = L&15.
            // All 16 center candidates for a row live in one 16-lane group.
            #pragma unroll
            for (int v = 0; v < 8; ++v) {
                const int t = t0 + v + mhi;
                float d2 = FLT_MAX;
                if (kk < KC && t < TN)
                    d2 = x2_s[t] + c2_s[kk] - 2.0f * acc[v];
                float bd = d2; int bk = kk;
                #pragma unroll
                for (int off = 8; off >= 1; off >>= 1) {
                    const float od = __shfl_xor(bd, off, 16);
                    const int   ok = __shfl_xor(bk, off, 16);
                    if (od < bd || (od == bd && ok < bk)) { bd = od; bk = ok; }
                }
                if (kk == 0 && t < TN) ids_s[t] = bk;
            }
        }
        __syncthreads();

        // ---- zero accumulators ----
        for (int i = tid; i < KC * HIDN; i += 256) sums_s[i] = 0.f;
        if (tid < 16) cnt_s[tid] = 0.f;
        __syncthreads();

        // ---- scatter: sums[k] += X[t], counts[k] += 1 (LDS float atomics) ----
        for (int i = tid; i < TN * HIDN; i += 256) {
            const int t = i >> 7;
            const int c = i & 127;
            atomicAdd(&sums_s[ids_s[t] * HIDN + c], Xs[i]);
        }
        if (tid < TN) atomicAdd(&cnt_s[ids_s[tid]], 1.0f);
        __syncthreads();

        // ---- center update: keep old center when count == 0 ----
        for (int i = tid; i < KC * HIDN; i += 256) {
            const float cnt = cnt_s[i >> 7];
            if (cnt > 0.f) Cs[i] = sums_s[i] / cnt;  // max(cnt,1)==cnt when >0
        }
        __syncthreads();
    }

    // ---- outputs: ids (as float) and centers ----
    if (tid < TN) out_ids[(size_t)b * TN + tid] = (float)ids_s[tid];
    for (int i = tid; i < KC * HIDN; i += 256)
        out_centers[(size_t)b * (KC * HIDN) + i] = Cs[i];
}

extern "C" void kernel_launch(void* const* d_in, const int* in_sizes, int n_in,
                              void* d_out, int out_size, void* d_ws, size_t ws_size,
                              hipStream_t stream) {
    const int*   user_ids = (const int*)d_in[0];
    const int*   seq      = (const int*)d_in[1];
    // d_in[2] pos_seqs, d_in[3] neg_seqs: unused by the reference computation
    const float* item_emb = (const float*)d_in[4];
    const float* user_emb = (const float*)d_in[5];
    const float* pos_emb  = (const float*)d_in[6];

    float* out_h       = (float*)d_out;
    float* out_ids     = out_h   + (size_t)BATCH * TN * HIDN;
    float* out_centers = out_ids + (size_t)BATCH * TN;

    const size_t smem_floats = (size_t)(TN * HIDN) + 16 * HIDN + KC * HIDN
                             + TN + 16 + 16 + TN;
    const size_t smem_bytes = smem_floats * sizeof(float);  // ~117.4 KB

    uptrec_kmeans_kernel<<<dim3(BATCH), dim3(256), smem_bytes, stream>>>(
        user_ids, seq, item_emb, user_emb, pos_emb,
        out_h, out_ids, out_centers);
}